// SpectralClusterer_57123065037312
// MI455X (gfx1250) — compile-verified
//
#include <hip/hip_runtime.h>
#include <math.h>

typedef float v2f __attribute__((ext_vector_type(2)));
typedef float v8f __attribute__((ext_vector_type(8)));

#define NN 2048
#define DD 512
#define BM 64
#define BN 128
#define BK 32
#define KPAD 36   // multiple of 4 -> 16B-aligned B128 async LDS writes; 36L mod 64 distinct -> conflict-free

// ---------------- q_i = sum_d w_d * x_id^2 ----------------
__global__ __launch_bounds__(256) void k_q(const float* __restrict__ x,
                                           const float* __restrict__ w,
                                           float* __restrict__ q) {
  __shared__ float red[256];
  const int i = blockIdx.x, t = threadIdx.x;
  float s = 0.f;
  for (int d = t; d < DD; d += 256) { float xv = x[(size_t)i * DD + d]; s += w[d] * xv * xv; }
  red[t] = s; __syncthreads();
  for (int o = 128; o > 0; o >>= 1) { if (t < o) red[t] += red[t + o]; __syncthreads(); }
  if (t == 0) q[i] = red[0];
}

// async copy of 16 bytes: global (base + voff) -> LDS (ldsoff); tracked by ASYNCcnt
__device__ __forceinline__ void async_g2l_b128(unsigned ldsoff, unsigned voff,
                                               const float* base) {
  asm volatile("global_load_async_to_lds_b128 %0, %1, %2"
               :
               : "v"(ldsoff), "v"(voff), "s"(base)
               : "memory");
}
__device__ __forceinline__ void wait_async0() {
  asm volatile("s_wait_asynccnt 0" ::: "memory");
}

// ------- fused WMMA GEMM + sigmoid: W = sigmoid(q_i + q_j - 2 * x diag(w) x^T + b) -------
__global__ __launch_bounds__(256) void k_gemm_sigmoid(const float* __restrict__ x,
                                                      const float* __restrict__ w,
                                                      const float* __restrict__ q,
                                                      const float* __restrict__ bptr,
                                                      float* __restrict__ Wout) {
  __shared__ float As[BM * KPAD];   // raw x rows of i-block
  __shared__ float Bs[BN * KPAD];   // raw x rows of j-block

  const int tid  = threadIdx.x;
  const int i0   = blockIdx.y * BM;
  const int j0   = blockIdx.x * BN;
  const int lane = tid & 31;
  const int wv   = tid >> 5;
  const int mw   = (wv & 1) * 32;   // wave M offset within block tile
  const int nw   = (wv >> 1) * 32;  // wave N offset within block tile
  const int half = lane >> 4;       // 0: K=k,k+1   1: K=k+2,k+3
  const int l15  = lane & 15;

  v8f acc[2][2] = {};

  // panel loader mapping: 16B (4 floats) per lane per instruction
  const int lr = tid >> 3;          // 0..31 (panel row)
  const int c4 = (tid & 7) * 4;     // 0..28 (col group of 4)

  // LDS byte offsets (low 32 bits of flat LDS address == ds offset)
  const unsigned asBase = (unsigned)(size_t)(&As[0]);
  const unsigned bsBase = (unsigned)(size_t)(&Bs[0]);

  for (int kc = 0; kc < DD; kc += BK) {
    __syncthreads();   // previous chunk's fragment reads done before overwriting panels
    // A panel: 64 rows x 32 cols of x (i-block), async global->LDS
#pragma unroll
    for (int rr = 0; rr < BM; rr += 32) {
      const unsigned voff = (unsigned)(((i0 + lr + rr) * DD + kc + c4) * 4);
      const unsigned loff = asBase + (unsigned)(((lr + rr) * KPAD + c4) * 4);
      async_g2l_b128(loff, voff, x);
    }
    // B panel: 128 rows x 32 cols of x (j-block), async global->LDS
#pragma unroll
    for (int rr = 0; rr < BN; rr += 32) {
      const unsigned voff = (unsigned)(((j0 + lr + rr) * DD + kc + c4) * 4);
      const unsigned loff = bsBase + (unsigned)(((lr + rr) * KPAD + c4) * 4);
      async_g2l_b128(loff, voff, x);
    }
    wait_async0();
    __syncthreads();

#pragma unroll
    for (int k = 0; k < BK; k += 4) {
      const int ko = k + 2 * half;                 // 8B-aligned (KPAD even, ko even)
      const v2f wk = *(const v2f*)(w + kc + ko);   // w for this fragment's two K values
      v2f a[2], b[2];
#pragma unroll
      for (int tm = 0; tm < 2; ++tm) {
        a[tm] = *(const v2f*)&As[(mw + tm * 16 + l15) * KPAD + ko];
        a[tm] = a[tm] * wk;                        // apply diag(w) along K to the A side
      }
#pragma unroll
      for (int tn = 0; tn < 2; ++tn)
        b[tn] = *(const v2f*)&Bs[(nw + tn * 16 + l15) * KPAD + ko];
#pragma unroll
      for (int tm = 0; tm < 2; ++tm)
#pragma unroll
        for (int tn = 0; tn < 2; ++tn)
          acc[tm][tn] = __builtin_amdgcn_wmma_f32_16x16x4_f32(
              false, a[tm], false, b[tn], (short)0, acc[tm][tn], false, false);
    }
  }

  // epilogue: s = q_i + q_j - 2*cross + b ; W = sigmoid(s)
  const float bb = bptr[0];
#pragma unroll
  for (int tm = 0; tm < 2; ++tm) {
#pragma unroll
    for (int tn = 0; tn < 2; ++tn) {
      const int   col = j0 + nw + tn * 16 + l15;
      const float qc  = q[col];
#pragma unroll
      for (int r = 0; r < 8; ++r) {
        const int   row = i0 + mw + tm * 16 + half * 8 + r;   // C/D: vgpr r -> M=r (lanes<16), r+8
        const float s   = q[row] + qc - 2.0f * acc[tm][tn][r] + bb;
        Wout[(size_t)row * NN + col] = 1.0f / (1.0f + __expf(-s));
      }
    }
  }
}

// ---------------- degree_i = sum_j W_ij ----------------
__global__ __launch_bounds__(256) void k_degree(const float* __restrict__ Wm,
                                                float* __restrict__ deg) {
  __shared__ float red[256];
  const int i = blockIdx.x, t = threadIdx.x;
  const float4* r4 = (const float4*)(Wm + (size_t)i * NN);
  float s = 0.f;
  for (int j = t; j < NN / 4; j += 256) { float4 a = r4[j]; s += a.x + a.y + a.z + a.w; }
  red[t] = s; __syncthreads();
  for (int o = 128; o > 0; o >>= 1) { if (t < o) red[t] += red[t + o]; __syncthreads(); }
  if (t == 0) deg[i] = red[0];
}

// ---------------- matvec: mode 0: y = L v ; mode 1: y = (sigma*I - L) v ----------------
__global__ __launch_bounds__(256) void k_mv(const float* __restrict__ Wm,
                                            const float* __restrict__ deg,
                                            const float* __restrict__ v,
                                            float* __restrict__ y,
                                            const float* __restrict__ scal, int mode) {
  __shared__ float red[256];
  const int i = blockIdx.x, t = threadIdx.x;
  const float4* r4 = (const float4*)(Wm + (size_t)i * NN);
  const float4* v4 = (const float4*)v;
  float s = 0.f;
  for (int j = t; j < NN / 4; j += 256) {
    float4 a = r4[j], b = v4[j];
    s += a.x * b.x + a.y * b.y + a.z * b.z + a.w * b.w;
  }
  red[t] = s; __syncthreads();
  for (int o = 128; o > 0; o >>= 1) { if (t < o) red[t] += red[t + o]; __syncthreads(); }
  if (t == 0) {
    const float wv = red[0];
    y[i] = mode ? (scal[0] - deg[i]) * v[i] + wv : deg[i] * v[i] - wv;
  }
}

// ---------------- postA: Rayleigh + normalize (lambda_max phase) ----------------
__global__ __launch_bounds__(1024) void k_postA(const float* __restrict__ y,
                                                float* __restrict__ v,
                                                float* __restrict__ scal) {
  __shared__ float r1[1024], r2[1024];
  __shared__ float invn;
  const int t = threadIdx.x;
  float sq = 0.f, dt = 0.f;
  for (int i = t; i < NN; i += 1024) { float yv = y[i]; sq += yv * yv; dt += v[i] * yv; }
  r1[t] = sq; r2[t] = dt; __syncthreads();
  for (int o = 512; o > 0; o >>= 1) {
    if (t < o) { r1[t] += r1[t + o]; r2[t] += r2[t + o]; }
    __syncthreads();
  }
  if (t == 0) { scal[1] = r2[0]; invn = 1.0f / (sqrtf(r1[0]) + 1e-30f); }
  __syncthreads();
  const float in_ = invn;
  for (int i = t; i < NN; i += 1024) v[i] = y[i] * in_;
}

__global__ void k_sigma(float* scal) { scal[0] = fabsf(scal[1]) * 1.02f + 5.0f; }

// ---------------- postB: deflate constant vector (mean-subtract) + normalize ----------------
__global__ __launch_bounds__(1024) void k_postB(const float* __restrict__ y,
                                                float* __restrict__ v) {
  __shared__ float r1[1024];
  __shared__ float mean_s, invn_s;
  const int t = threadIdx.x;
  float s = 0.f;
  for (int i = t; i < NN; i += 1024) s += y[i];
  r1[t] = s; __syncthreads();
  for (int o = 512; o > 0; o >>= 1) { if (t < o) r1[t] += r1[t + o]; __syncthreads(); }
  if (t == 0) mean_s = r1[0] * (1.0f / NN);
  __syncthreads();
  const float mu = mean_s;
  float sq = 0.f;
  for (int i = t; i < NN; i += 1024) { float z = y[i] - mu; sq += z * z; }
  __syncthreads();
  r1[t] = sq; __syncthreads();
  for (int o = 512; o > 0; o >>= 1) { if (t < o) r1[t] += r1[t + o]; __syncthreads(); }
  if (t == 0) invn_s = 1.0f / (sqrtf(r1[0]) + 1e-30f);
  __syncthreads();
  const float inv = invn_s;
  for (int i = t; i < NN; i += 1024) v[i] = (y[i] - mu) * inv;
}

// ---------------- postC: fiedler value = sigma - v.(Mv); emit value + vector ----------------
__global__ __launch_bounds__(1024) void k_postC(const float* __restrict__ v,
                                                const float* __restrict__ y,
                                                const float* __restrict__ scal,
                                                float* __restrict__ outVal,
                                                float* __restrict__ outVec) {
  __shared__ float r1[1024];
  const int t = threadIdx.x;
  float dt = 0.f;
  for (int i = t; i < NN; i += 1024) dt += v[i] * y[i];
  r1[t] = dt; __syncthreads();
  for (int o = 512; o > 0; o >>= 1) { if (t < o) r1[t] += r1[t + o]; __syncthreads(); }
  if (t == 0) outVal[0] = scal[0] - r1[0];
  for (int i = t; i < NN; i += 1024) outVec[i] = v[i];
}

// ---------------- deterministic pseudo-random init ----------------
__global__ void k_init_v(float* __restrict__ v, unsigned seed) {
  const int i = blockIdx.x * blockDim.x + threadIdx.x;
  if (i < NN) {
    unsigned u = (unsigned)i * 2654435761u + seed;
    u ^= u >> 16; u *= 2246822519u; u ^= u >> 13; u *= 3266489917u; u ^= u >> 16;
    v[i] = ((float)(u & 0xFFFFFFu) / 8388608.0f) - 1.0f;
  }
}

extern "C" void kernel_launch(void* const* d_in, const int* in_sizes, int n_in,
                              void* d_out, int out_size, void* d_ws, size_t ws_size,
                              hipStream_t stream) {
  const float* x = (const float*)d_in[0];   // (2048, 512)
  const float* w = (const float*)d_in[1];   // (512,)
  const float* b = (const float*)d_in[2];   // (1,)

  float* Wout   = (float*)d_out;                 // 2048*2048 grouping matrix
  float* outVal = Wout + (size_t)NN * NN;        // fiedler value
  float* outVec = outVal + 1;                    // fiedler vector (2048)

  float* ws   = (float*)d_ws;
  float* q    = ws;             // 2048
  float* deg  = ws + NN;        // 2048
  float* v    = ws + 2 * NN;    // 2048
  float* y    = ws + 3 * NN;    // 2048
  float* scal = ws + 4 * NN;    // [0]=sigma [1]=rayleigh

  // 1) q, 2) fused WMMA GEMM + sigmoid -> W, 3) degrees
  k_q<<<NN, 256, 0, stream>>>(x, w, q);
  dim3 grid(NN / BN, NN / BM);
  k_gemm_sigmoid<<<grid, 256, 0, stream>>>(x, w, q, b, Wout);
  k_degree<<<NN, 256, 0, stream>>>(Wout, deg);

  // 4) estimate lambda_max(L) by power iteration
  k_init_v<<<NN / 256, 256, 0, stream>>>(v, 12345u);
  for (int it = 0; it < 40; ++it) {
    k_mv<<<NN, 256, 0, stream>>>(Wout, deg, v, y, scal, 0);
    k_postA<<<1, 1024, 0, stream>>>(y, v, scal);
  }
  k_sigma<<<1, 1, 0, stream>>>(scal);

  // 5) deflated power iteration on sigma*I - L -> Fiedler vector
  k_init_v<<<NN / 256, 256, 0, stream>>>(v, 777u);
  for (int it = 0; it < 1000; ++it) {
    k_mv<<<NN, 256, 0, stream>>>(Wout, deg, v, y, scal, 1);
    k_postB<<<1, 1024, 0, stream>>>(y, v);
  }

  // 6) Fiedler value = sigma - v^T (sigma*I - L) v ; emit outputs
  k_mv<<<NN, 256, 0, stream>>>(Wout, deg, v, y, scal, 1);
  k_postC<<<1, 1024, 0, stream>>>(v, y, scal, outVal, outVec);
}